// TrendLoss_42305427865960
// MI455X (gfx1250) — compile-verified
//
#include <hip/hip_runtime.h>
#include <stdint.h>

#define DAYS 365
#define YRS  30
#define NS   4000
#define PAD  384           // 365 padded to 6*64 for WMMA tiles
#define NPAIR 435          // C(30,2)
#define MEDR 217           // (435-1)/2
#define NTOT 43800000      // T*S = 10950*4000

typedef float v2f __attribute__((ext_vector_type(2)));
typedef float v8f __attribute__((ext_vector_type(8)));

#if defined(__has_builtin)
#if __has_builtin(__builtin_amdgcn_wmma_f32_16x16x4_f32)
#define HAVE_WMMA4 1
#endif
#endif

// ---------------- init: zero the two accumulators ----------------
__global__ void k_init(float* ws) {
    if (threadIdx.x < 2) ws[threadIdx.x] = 0.0f;
}

// ---------------- stage A: sum of squared diffs ----------------
__global__ __launch_bounds__(256) void k_rmse(const float* __restrict__ pred,
                                              const float* __restrict__ obs,
                                              float* __restrict__ ssq, int n) {
    int tid = blockIdx.x * blockDim.x + threadIdx.x;
    int stride = gridDim.x * blockDim.x;
    float acc = 0.0f;
    for (int i = tid; i < n; i += stride) {
        float d = pred[i] - obs[i];
        acc += d * d;
    }
    for (int o = 16; o > 0; o >>= 1) acc += __shfl_xor(acc, o, 32);
    __shared__ float parts[8];
    int lane = threadIdx.x & 31, w = threadIdx.x >> 5;
    if (lane == 0) parts[w] = acc;
    __syncthreads();
    if (threadIdx.x == 0) {
        float s = 0.0f;
        for (int i = 0; i < 8; ++i) s += parts[i];
        atomicAdd(ssq, s);
    }
}

// ---------------- stage B: per (station,year) mean + 5 order stats ----------------
// One wave handles one (station,year). Day data gathered into LDS with CDNA5
// async global->LDS loads; mean via v_wmma_f32_16x16x4_f32 with all-ones B;
// order statistics via rank counting (ties broken by index).
__global__ __launch_bounds__(256) void k_stats(const float* __restrict__ pred,
                                               const float* __restrict__ obs,
                                               float* __restrict__ pM, float* __restrict__ tM,
                                               float* __restrict__ pQ, float* __restrict__ tQ) {
    __shared__ float buf[8][PAD];
    __shared__ float qs[8][5];
    const int w = threadIdx.x >> 5, lane = threadIdx.x & 31;
    const int pair = blockIdx.x * 8 + w;       // 120000 total, 15000 blocks exactly
    const int y = pair / NS;
    const int s = pair - y * NS;
    const int qr0 = 364, qr1 = 357, qr2 = 182, qr3 = 109, qr4 = 7;

    for (int arr = 0; arr < 2; ++arr) {
        const float* src = arr ? obs : pred;

        // make sure prior-iteration LDS reads are done before async engine rewrites buf
        asm volatile("s_wait_dscnt 0x0" ::: "memory");

        // async gather: 384 per-wave values, day stride = NS floats (16 KB)
        #pragma unroll
        for (int r = 0; r < 12; ++r) {
            int d = lane + 32 * r;
            int dd = d > 364 ? 364 : d;       // clamp; padding overwritten below
            const float* gp = src + (size_t)(y * DAYS + dd) * NS + s;
            unsigned la = (unsigned)(uintptr_t)(&buf[w][d]);
            unsigned long long ga = (unsigned long long)(uintptr_t)gp;
            asm volatile("global_load_async_to_lds_b32 %0, %1, off"
                         :: "v"(la), "v"(ga) : "memory");
        }
        asm volatile("s_wait_asynccnt 0x0" ::: "memory");
        if (lane < PAD - DAYS) buf[w][DAYS + lane] = 0.0f;   // zero pad (exact for sum)
        __syncthreads();

        // ---- mean of 365 values ----
        float total;
#ifdef HAVE_WMMA4
        {
            // B = all-ones 4x16  =>  D[m][n] = sum_k A[m][k]; summing a column over
            // both lane-halves gives the tile total, so A slot assignment is free.
            v8f acc = {0.f,0.f,0.f,0.f,0.f,0.f,0.f,0.f};
            v2f ones = {1.0f, 1.0f};
            #pragma unroll
            for (int c = 0; c < 6; ++c) {
                v2f a = { buf[w][c * 64 + lane], buf[w][c * 64 + 32 + lane] };
                acc = __builtin_amdgcn_wmma_f32_16x16x4_f32(
                        false, a, false, ones, (short)0, acc, false, false);
            }
            float colsum = acc[0]+acc[1]+acc[2]+acc[3]+acc[4]+acc[5]+acc[6]+acc[7];
            total = colsum + __shfl_xor(colsum, 16, 32);   // combine M halves
        }
#else
        {
            float psum = 0.0f;
            #pragma unroll
            for (int r = 0; r < 12; ++r) psum += buf[w][lane + 32 * r];
            for (int o = 16; o > 0; o >>= 1) psum += __shfl_xor(psum, o, 32);
            total = psum;
        }
#endif
        float meanv = total * (1.0f / 365.0f);

        // ---- order statistics via rank counting ----
        for (int r = 0; r < 12; ++r) {
            int e = lane + 32 * r;
            if (e < DAYS) {
                float x = buf[w][e];
                int rank = 0;
                for (int j = 0; j < DAYS; ++j) {
                    float xj = buf[w][j];
                    rank += (xj < x) || (xj == x && j < e);
                }
                if (rank == qr0) qs[w][0] = x;
                if (rank == qr1) qs[w][1] = x;
                if (rank == qr2) qs[w][2] = x;
                if (rank == qr3) qs[w][3] = x;
                if (rank == qr4) qs[w][4] = x;
            }
        }
        __syncthreads();

        float* M  = arr ? tM : pM;
        float* Qo = arr ? tQ : pQ;
        if (lane == 0) M[s * YRS + y] = meanv;
        if (lane < 5)  Qo[(s * 5 + lane) * YRS + y] = qs[w][lane];
        __syncthreads();
    }
}

// ---------------- stage C: per-station Theil-Sen medians + loss terms ----------------
__global__ __launch_bounds__(256) void k_trend(const float* __restrict__ pM,
                                               const float* __restrict__ tM,
                                               const float* __restrict__ pQ,
                                               const float* __restrict__ tQ,
                                               float* __restrict__ accum) {
    __shared__ float sl[8][NPAIR + 1];
    __shared__ float med[8][12];
    const int w = threadIdx.x >> 5, lane = threadIdx.x & 31;
    const int s = blockIdx.x * 8 + w;          // 4000 stations, 500 blocks exactly

    for (int k = 0; k < 12; ++k) {
        const float* rowp;
        if (k == 0)       rowp = pM + s * YRS;
        else if (k == 1)  rowp = tM + s * YRS;
        else {
            int q = (k - 2) >> 1, isT = (k - 2) & 1;
            rowp = (isT ? tQ : pQ) + (s * 5 + q) * YRS;
        }
        float xv = (lane < YRS) ? rowp[lane] : 0.0f;

        // pairwise slopes (shuffles hoisted out of divergence: all 32 lanes shuffle)
        for (int r = 0; r < 14; ++r) {
            int p = lane + 32 * r;
            int pc = p < NPAIR ? p : NPAIR - 1;
            int pp = pc, i = 0;
            while (pp >= (YRS - 1 - i)) { pp -= (YRS - 1 - i); ++i; }
            int j = i + 1 + pp;
            float xi = __shfl(xv, i, 32);
            float xj = __shfl(xv, j, 32);
            float slope = (xj - xi) / (float)(j - i);
            if (p < NPAIR) sl[w][p] = slope;
        }
        __syncthreads();

        // median = rank 217 of 435 (ties broken by index => unique)
        for (int r = 0; r < 14; ++r) {
            int e = lane + 32 * r;
            if (e < NPAIR) {
                float x = sl[w][e];
                int rank = 0;
                for (int jj = 0; jj < NPAIR; ++jj) {
                    float xj = sl[w][jj];
                    rank += (xj < x) || (xj == x && jj < e);
                }
                if (rank == MEDR) med[w][k] = x;
            }
        }
        __syncthreads();
    }

    if (lane == 0) {
        float mp = med[w][0], mt = med[w][1];
        float res = (mt - mp) * (mt - mp);
        #pragma unroll
        for (int q = 0; q < 5; ++q) {
            float a = med[w][2 + q * 2];   // slope_pQ
            float b = med[w][3 + q * 2];   // slope_tQ
            float ratio = b / (-a);
            res += ratio * ratio;
        }
        atomicAdd(accum, res);
    }
}

// ---------------- final combine ----------------
__global__ void k_final(const float* __restrict__ ws, float* __restrict__ out) {
    if (threadIdx.x == 0 && blockIdx.x == 0) {
        float rmse = sqrtf(ws[0] / (float)NTOT);
        out[0] = rmse + ws[1] / (float)NS;
    }
}

extern "C" void kernel_launch(void* const* d_in, const int* in_sizes, int n_in,
                              void* d_out, int out_size, void* d_ws, size_t ws_size,
                              hipStream_t stream) {
    (void)in_sizes; (void)n_in; (void)out_size; (void)ws_size;
    const float* pred = (const float*)d_in[0];
    const float* obs  = (const float*)d_in[1];
    float* wsf = (float*)d_ws;

    float* ssq   = wsf + 0;
    float* trend = wsf + 1;
    float* pM = wsf + 16;                  // S*YRS = 120000
    float* tM = pM + NS * YRS;             // 120000
    float* pQ = tM + NS * YRS;             // S*5*YRS = 600000
    float* tQ = pQ + NS * 5 * YRS;         // 600000   (total ~5.8 MB)

    k_init <<<1, 32, 0, stream>>>(wsf);
    k_rmse <<<4096, 256, 0, stream>>>(pred, obs, ssq, NTOT);
    k_stats<<<(NS * YRS) / 8, 256, 0, stream>>>(pred, obs, pM, tM, pQ, tQ);
    k_trend<<<NS / 8, 256, 0, stream>>>(pM, tM, pQ, tQ, trend);
    k_final<<<1, 1, 0, stream>>>(wsf, (float*)d_out);
}